// Net_46875273068791
// MI455X (gfx1250) — compile-verified
//
#include <hip/hip_runtime.h>
#include <math.h>

typedef __attribute__((ext_vector_type(2))) float v2f;
typedef __attribute__((ext_vector_type(8))) float v8f;

#define NN 100000
#define NE 6400000
#define NTILES (NN / 16)   // 6250 exactly -> no tail guard needed for nodes

// ---------------------------------------------------------------------------
// Layer-1 node transform via f32 WMMA:  H0 = X@w1[0], H1 = X@w1[1],
// BASE1 = X@root1 + b1.   X: (NN,5), weights: (5,16).
// One wave32 per 16-node tile. K=5 split into two 16x16x4 WMMA steps.
// ---------------------------------------------------------------------------
__global__ void node_xform1(const float* __restrict__ X,
                            const float* __restrict__ W1,   // (2,5,16)
                            const float* __restrict__ R1,   // (5,16)
                            const float* __restrict__ B1,   // (16)
                            float* __restrict__ H0,
                            float* __restrict__ H1,
                            float* __restrict__ BASE1) {
  int wv   = (blockIdx.x * blockDim.x + threadIdx.x) >> 5;  // wave-uniform
  int lane = threadIdx.x & 31;
  if (wv >= NTILES) return;                                  // uniform branch
  int m    = lane & 15;
  int half = lane >> 4;            // 0: K=k0,k0+1  1: K=k0+2,k0+3
  int node = wv * 16 + m;
  const float* xr = X + node * 5;

  // A tile (16x4 per chunk): lane holds row m, two K values.
  int ka = 2 * half;
  v2f a0; a0.x = xr[ka]; a0.y = xr[ka + 1];                  // K=0..3
  v2f a1; a1.x = (half == 0) ? xr[4] : 0.0f; a1.y = 0.0f;    // K=4 (5..7 pad)

  int n = m;                        // output column for B/C/D
  const float* Wm[3] = {W1, W1 + 80, R1};
  v8f acc[3];
  #pragma unroll
  for (int t = 0; t < 3; ++t) {
    const float* W = Wm[t];
    int kk = 2 * half;
    v2f b0; b0.x = W[kk * 16 + n]; b0.y = W[(kk + 1) * 16 + n];          // rows 0..3
    v2f b1; b1.x = (half == 0) ? W[4 * 16 + n] : 0.0f; b1.y = 0.0f;      // row 4
    v8f c = {};
    c = __builtin_amdgcn_wmma_f32_16x16x4_f32(false, a0, false, b0, (short)0, c, false, false);
    c = __builtin_amdgcn_wmma_f32_16x16x4_f32(false, a1, false, b1, (short)0, c, false, false);
    acc[t] = c;
  }
  float bias = B1[n];
  #pragma unroll
  for (int r = 0; r < 8; ++r) {
    int mm = r + 8 * half;
    int nd = wv * 16 + mm;
    H0[nd * 16 + n]    = acc[0][r];
    H1[nd * 16 + n]    = acc[1][r];
    BASE1[nd * 16 + n] = acc[2][r] + bias;
  }
}

// ---------------------------------------------------------------------------
// Layer-2 node transform via f32 WMMA: G0 = H@w2[0], G1 = H@w2[1],
// BASE2 = H@root2 + b2.  H: (NN,16), weights (16,4) padded to N=16.
// ---------------------------------------------------------------------------
__global__ void node_xform2(const float* __restrict__ H,
                            const float* __restrict__ W2,   // (2,16,4)
                            const float* __restrict__ R2,   // (16,4)
                            const float* __restrict__ B2,   // (4)
                            float* __restrict__ G0,
                            float* __restrict__ G1,
                            float* __restrict__ BASE2) {
  int wv   = (blockIdx.x * blockDim.x + threadIdx.x) >> 5;
  int lane = threadIdx.x & 31;
  if (wv >= NTILES) return;
  int m    = lane & 15;
  int half = lane >> 4;
  int node = wv * 16 + m;
  const float* hr = H + node * 16;
  int n = m;
  bool ncol = (n < 4);
  const float* Wm[3] = {W2, W2 + 64, R2};
  v8f acc[3] = {};
  #pragma unroll
  for (int c = 0; c < 4; ++c) {     // K = 16 -> 4 chunks of 4
    int kk = 4 * c + 2 * half;
    v2f a; a.x = hr[kk]; a.y = hr[kk + 1];
    #pragma unroll
    for (int t = 0; t < 3; ++t) {
      const float* W = Wm[t];
      v2f b;
      b.x = ncol ? W[kk * 4 + n] : 0.0f;
      b.y = ncol ? W[(kk + 1) * 4 + n] : 0.0f;
      acc[t] = __builtin_amdgcn_wmma_f32_16x16x4_f32(false, a, false, b, (short)0, acc[t], false, false);
    }
  }
  if (ncol) {
    float bias = B2[n];
    #pragma unroll
    for (int r = 0; r < 8; ++r) {
      int mm = r + 8 * half;
      int nd = wv * 16 + mm;
      G0[nd * 4 + n]    = acc[0][r];
      G1[nd * 4 + n]    = acc[1][r];
      BASE2[nd * 4 + n] = acc[2][r] + bias;
    }
  }
}

// ---------------------------------------------------------------------------
// Edge scatter, layer 1: msg = (1-p)*H0[src] + p*H1[src]; atomic add to AGG1[dst].
// Also accumulates in-degree (shared by both layers).
// ---------------------------------------------------------------------------
__global__ void edge_scatter1(const long long* __restrict__ SRC,
                              const long long* __restrict__ DST,
                              const float* __restrict__ P,
                              const float* __restrict__ H0,
                              const float* __restrict__ H1,
                              float* __restrict__ AGG1,
                              float* __restrict__ CNT) {
  int e = blockIdx.x * blockDim.x + threadIdx.x;
  if (e >= NE) return;
  int s = (int)SRC[e];
  int d = (int)DST[e];
  float p  = P[e];
  float w0 = 1.0f - p;
  const float4* a = (const float4*)(H0 + (size_t)s * 16);
  const float4* b = (const float4*)(H1 + (size_t)s * 16);
  float* out = AGG1 + (size_t)d * 16;
  #pragma unroll
  for (int j = 0; j < 4; ++j) {
    float4 va = a[j], vb = b[j];
    atomicAdd(out + j * 4 + 0, w0 * va.x + p * vb.x);
    atomicAdd(out + j * 4 + 1, w0 * va.y + p * vb.y);
    atomicAdd(out + j * 4 + 2, w0 * va.z + p * vb.z);
    atomicAdd(out + j * 4 + 3, w0 * va.w + p * vb.w);
  }
  atomicAdd(CNT + d, 1.0f);
}

// ---------------------------------------------------------------------------
// Edge scatter, layer 2 (4 channels).
// ---------------------------------------------------------------------------
__global__ void edge_scatter2(const long long* __restrict__ SRC,
                              const long long* __restrict__ DST,
                              const float* __restrict__ P,
                              const float* __restrict__ G0,
                              const float* __restrict__ G1,
                              float* __restrict__ AGG2) {
  int e = blockIdx.x * blockDim.x + threadIdx.x;
  if (e >= NE) return;
  int s = (int)SRC[e];
  int d = (int)DST[e];
  float p  = P[e];
  float w0 = 1.0f - p;
  float4 va = *(const float4*)(G0 + (size_t)s * 4);
  float4 vb = *(const float4*)(G1 + (size_t)s * 4);
  float* out = AGG2 + (size_t)d * 4;
  atomicAdd(out + 0, w0 * va.x + p * vb.x);
  atomicAdd(out + 1, w0 * va.y + p * vb.y);
  atomicAdd(out + 2, w0 * va.z + p * vb.z);
  atomicAdd(out + 3, w0 * va.w + p * vb.w);
}

// ---------------------------------------------------------------------------
// h = elu(AGG1 / max(cnt,1) + BASE1)
// ---------------------------------------------------------------------------
__global__ void finalize1(const float* __restrict__ AGG1,
                          const float* __restrict__ CNT,
                          const float* __restrict__ BASE1,
                          float* __restrict__ H) {
  int i = blockIdx.x * blockDim.x + threadIdx.x;
  if (i >= NN * 16) return;
  int node = i >> 4;
  float c = fmaxf(CNT[node], 1.0f);
  float v = AGG1[i] / c + BASE1[i];
  H[i] = (v > 0.0f) ? v : expm1f(v);
}

// ---------------------------------------------------------------------------
// out = log_softmax(AGG2 / max(cnt,1) + BASE2, axis=1)   (4 channels)
// ---------------------------------------------------------------------------
__global__ void final2(const float* __restrict__ AGG2,
                       const float* __restrict__ CNT,
                       const float* __restrict__ BASE2,
                       float* __restrict__ OUT) {
  int node = blockIdx.x * blockDim.x + threadIdx.x;
  if (node >= NN) return;
  float c = fmaxf(CNT[node], 1.0f);
  float v[4];
  #pragma unroll
  for (int j = 0; j < 4; ++j)
    v[j] = AGG2[node * 4 + j] / c + BASE2[node * 4 + j];
  float mx = fmaxf(fmaxf(v[0], v[1]), fmaxf(v[2], v[3]));
  float s = 0.0f;
  #pragma unroll
  for (int j = 0; j < 4; ++j) s += expf(v[j] - mx);
  float l = mx + logf(s);
  #pragma unroll
  for (int j = 0; j < 4; ++j)
    OUT[node * 4 + j] = v[j] - l;
}

__global__ void zero_kernel(float* __restrict__ p, int n) {
  int i = blockIdx.x * blockDim.x + threadIdx.x;
  if (i < n) p[i] = 0.0f;
}

extern "C" void kernel_launch(void* const* d_in, const int* in_sizes, int n_in,
                              void* d_out, int out_size, void* d_ws, size_t ws_size,
                              hipStream_t stream) {
  const float*     X  = (const float*)d_in[0];
  const long long* EI = (const long long*)d_in[1];
  const float*     P  = (const float*)d_in[2];
  const float*     W1 = (const float*)d_in[3];
  const float*     R1 = (const float*)d_in[4];
  const float*     B1 = (const float*)d_in[5];
  const float*     W2 = (const float*)d_in[6];
  const float*     R2 = (const float*)d_in[7];
  const float*     B2 = (const float*)d_in[8];
  float* OUT = (float*)d_out;

  float* ws    = (float*)d_ws;
  float* H0    = ws;                 // NN*16
  float* H1    = H0 + 1600000;       // NN*16
  float* BASE1 = H1 + 1600000;       // NN*16
  float* H     = BASE1 + 1600000;    // NN*16
  float* G0    = H + 1600000;        // NN*4
  float* G1    = G0 + 400000;        // NN*4
  float* BASE2 = G1 + 400000;        // NN*4
  float* AGG1  = BASE2 + 400000;     // NN*16  (zeroed)
  float* AGG2  = AGG1 + 1600000;     // NN*4   (zeroed)
  float* CNT   = AGG2 + 400000;      // NN     (zeroed)

  const long long* SRC = EI;
  const long long* DST = EI + NE;

  int zn = 1600000 + 400000 + 100000;   // AGG1 + AGG2 + CNT contiguous
  zero_kernel<<<(zn + 255) / 256, 256, 0, stream>>>(AGG1, zn);

  node_xform1<<<(NTILES * 32 + 255) / 256, 256, 0, stream>>>(X, W1, R1, B1, H0, H1, BASE1);
  edge_scatter1<<<(NE + 255) / 256, 256, 0, stream>>>(SRC, DST, P, H0, H1, AGG1, CNT);
  finalize1<<<(NN * 16 + 255) / 256, 256, 0, stream>>>(AGG1, CNT, BASE1, H);
  node_xform2<<<(NTILES * 32 + 255) / 256, 256, 0, stream>>>(H, W2, R2, B2, G0, G1, BASE2);
  edge_scatter2<<<(NE + 255) / 256, 256, 0, stream>>>(SRC, DST, P, G0, G1, AGG2);
  final2<<<(NN + 255) / 256, 256, 0, stream>>>(AGG2, CNT, BASE2, OUT);
}